// TravelLoss_33380485825106
// MI455X (gfx1250) — compile-verified
//
#include <hip/hip_runtime.h>

typedef __attribute__((ext_vector_type(2))) float v2f;
typedef __attribute__((ext_vector_type(4))) float v4f;
typedef __attribute__((ext_vector_type(8))) float v8f;

#define NS      128                 // samples
#define DF      8128                // feature dim
#define NP      8128                // pairs = NS*(NS-1)/2
#define GELEMS  (NS * NS)           // Gram matrix elements
#define NB2     2048                // blocks in main reduction
#define EPSV    1e-12f

// Workspace layout (floats):
//   [0, 16384)          G_real (128x128)
//   [16384, 32768)      G_fake (128x128)
//   [32768, 40896)      inv_nr[NP]
//   [40896, 49024)      inv_nf[NP]
//   then int II[NP], int JJ[NP], float partials[NB2]

// ---------------------------------------------------------------------------
// Kernel 1: Gram matrices G = S * S^T via V_WMMA_F32_16X16X4_F32.
// One wave32 per 16x16 output tile; blockIdx encodes {matrix, tile}.
// A/B per-lane layout (ISA 7.12.2, 32-bit 16x4): lane L holds K = 2*(L>>4)+v
// for VGPR v -> a contiguous 8-byte (v2f) load per operand per K-step.
// ---------------------------------------------------------------------------
__global__ void gram_wmma(const float* __restrict__ Sr,
                          const float* __restrict__ Sf,
                          float* __restrict__ W) {
    const int mat  = blockIdx.x >> 6;      // 0 = real, 1 = fake
    const int tile = blockIdx.x & 63;      // 8x8 tiles of 16x16
    const int bi   = tile >> 3;
    const int bj   = tile & 7;

    const float* __restrict__ S = mat ? Sf : Sr;
    float* __restrict__ G = W + mat * GELEMS;

    const int lane = threadIdx.x;          // 0..31 (wave32)
    const int l16  = lane & 15;
    const int hi   = lane >> 4;

    const float* rowA = S + (size_t)(bi * 16 + l16) * DF + 2 * hi;
    const float* rowB = S + (size_t)(bj * 16 + l16) * DF + 2 * hi;

    v8f acc = {};
    for (int k0 = 0; k0 < DF; k0 += 4) {
        v2f a = *(const v2f*)(rowA + k0);
        v2f b = *(const v2f*)(rowB + k0);
        acc = __builtin_amdgcn_wmma_f32_16x16x4_f32(
            /*neg_a=*/false, a, /*neg_b=*/false, b,
            /*c_mod=*/(short)0, acc, /*reuse_a=*/false, /*reuse_b=*/false);
    }

    // C/D layout: VGPR r -> M = r + 8*hi, N = l16
    #pragma unroll
    for (int r = 0; r < 8; ++r) {
        G[(bi * 16 + r + 8 * hi) * NS + (bj * 16 + l16)] = acc[r];
    }
}

// ---------------------------------------------------------------------------
// Kernel 2: per-pair inverse norms from the Gram matrices + pair index tables.
// tril_indices(k=-1) order: p = i*(i-1)/2 + j,  j < i.
// ---------------------------------------------------------------------------
__global__ void pair_norms(float* __restrict__ W) {
    int p = blockIdx.x * blockDim.x + threadIdx.x;
    if (p >= NP) return;

    int i = (int)((1.0f + sqrtf(1.0f + 8.0f * (float)p)) * 0.5f);
    while (i * (i - 1) / 2 > p) --i;
    while ((i + 1) * i / 2 <= p) ++i;
    int j = p - i * (i - 1) / 2;

    const float* Gr = W;
    const float* Gf = W + GELEMS;
    float nr2 = Gr[i * NS + i] + Gr[j * NS + j] - 2.0f * Gr[i * NS + j];
    float nf2 = Gf[i * NS + i] + Gf[j * NS + j] - 2.0f * Gf[i * NS + j];

    W[2 * GELEMS + p]      = rsqrtf(fmaxf(nr2, 1e-30f));   // inv_nr
    W[2 * GELEMS + NP + p] = rsqrtf(fmaxf(nf2, 1e-30f));   // inv_nf

    int* II = (int*)(W + 2 * GELEMS + 2 * NP);
    int* JJ = II + NP;
    II[p] = i;
    JJ[p] = j;
}

// ---------------------------------------------------------------------------
// Kernel 3: total = sum_{p,q} max(vf(p,q)/nf[p], eps) * max(vr(q,p)/nr[q], eps)
// Strips of 8 contiguous q per p; deterministic fixed-order reductions.
// ---------------------------------------------------------------------------
__global__ void trace_reduce(const float* __restrict__ Sr,
                             const float* __restrict__ Sf,
                             const float* __restrict__ W,
                             float* __restrict__ partials) {
    const float* inv_nr = W + 2 * GELEMS;
    const float* inv_nf = inv_nr + NP;
    const int*   II     = (const int*)(inv_nf + NP);
    const int*   JJ     = II + NP;

    const int QCH    = DF / 8;        // 1016 strips per row
    const int STRIPS = NP * QCH;      // 8,258,048

    float acc = 0.0f;
    for (int s = blockIdx.x * blockDim.x + threadIdx.x; s < STRIPS;
         s += gridDim.x * blockDim.x) {
        int p  = s / QCH;
        int q0 = (s - p * QCH) * 8;

        const float* fa = Sf + (size_t)II[p] * DF + q0;
        const float* fb = Sf + (size_t)JJ[p] * DF + q0;
        const float invf = inv_nf[p];

        v4f a0 = *(const v4f*)(fa);
        v4f a1 = *(const v4f*)(fa + 4);
        v4f b0 = *(const v4f*)(fb);
        v4f b1 = *(const v4f*)(fb + 4);

        #pragma unroll
        for (int t = 0; t < 8; ++t) {
            int q = q0 + t;
            float va = (t < 4) ? a0[t] : a1[t - 4];
            float vb = (t < 4) ? b0[t] : b1[t - 4];
            float mf = fmaxf((va - vb) * invf, EPSV);
            float vr = (Sr[(size_t)II[q] * DF + p] - Sr[(size_t)JJ[q] * DF + p])
                       * inv_nr[q];
            float mr = fmaxf(vr, EPSV);
            acc = fmaf(mf, mr, acc);
        }
    }

    __shared__ float sd[256];
    sd[threadIdx.x] = acc;
    __syncthreads();
    for (int off = 128; off > 0; off >>= 1) {
        if (threadIdx.x < off) sd[threadIdx.x] += sd[threadIdx.x + off];
        __syncthreads();
    }
    if (threadIdx.x == 0) partials[blockIdx.x] = sd[0];
}

// ---------------------------------------------------------------------------
// Kernel 4: deterministic final reduction of the 2048 block partials.
// ---------------------------------------------------------------------------
__global__ void final_sum(const float* __restrict__ partials,
                          float* __restrict__ out) {
    __shared__ float sd[256];
    float acc = 0.0f;
    for (int i = threadIdx.x; i < NB2; i += 256) acc += partials[i];
    sd[threadIdx.x] = acc;
    __syncthreads();
    for (int off = 128; off > 0; off >>= 1) {
        if (threadIdx.x < off) sd[threadIdx.x] += sd[threadIdx.x + off];
        __syncthreads();
    }
    if (threadIdx.x == 0) out[0] = sd[0];
}

extern "C" void kernel_launch(void* const* d_in, const int* in_sizes, int n_in,
                              void* d_out, int out_size, void* d_ws, size_t ws_size,
                              hipStream_t stream) {
    const float* Sr = (const float*)d_in[0];   // S_real [128 x 8128] fp32
    const float* Sf = (const float*)d_in[1];   // S_fake [128 x 8128] fp32
    float* W = (float*)d_ws;
    float* partials = (float*)((int*)(W + 2 * GELEMS + 2 * NP) + 2 * NP);

    // 1) Gram matrices via f32 WMMA: 2 matrices x 64 tiles, one wave32 each.
    gram_wmma<<<128, 32, 0, stream>>>(Sr, Sf, W);
    // 2) Pair norms + index tables.
    pair_norms<<<(NP + 255) / 256, 256, 0, stream>>>(W);
    // 3) Main fused elementwise trace reduction.
    trace_reduce<<<NB2, 256, 0, stream>>>(Sr, Sf, W, partials);
    // 4) Deterministic scalar finish.
    final_sum<<<1, 256, 0, stream>>>(partials, (float*)d_out);
}